// Reconstruct_36653250904488
// MI455X (gfx1250) — compile-verified
//
#include <hip/hip_runtime.h>
#include <stdint.h>

// ---------------------------------------------------------------------------
// Problem constants (from reference): B=2, C=4, N=2^21, V=256^3
// out[k*V + v] = feat[k*N + idx[v]],  k = b*C+c in [0,8)
// ---------------------------------------------------------------------------
#define NPART 2097152
#define BC    8
#define VTOT  16777216

typedef float     v2f  __attribute__((ext_vector_type(2)));
typedef float     v4f  __attribute__((ext_vector_type(4)));
typedef long long v2ll __attribute__((ext_vector_type(2)));
typedef int       v4i  __attribute__((ext_vector_type(4)));

// CDNA5 async global->LDS copy path (probe via __has_builtin; device pass only)
#if defined(__gfx1250__) &&                                              \
    __has_builtin(__builtin_amdgcn_global_load_async_to_lds_b128) &&     \
    __has_builtin(__builtin_amdgcn_s_wait_asynccnt)
#define USE_ASYNC_LDS 1
typedef __attribute__((address_space(1))) v4i as1_v4i;  // global (AS1)
typedef __attribute__((address_space(3))) v4i as3_v4i;  // LDS (AS3)
#else
#define USE_ASYNC_LDS 0
#endif

// ---------------------------------------------------------------------------
// Pass 1: pack feat (8, N) -> packed (N, 8) so each particle's 8 channel
// values are one contiguous 32B record (L2-resident gather table).
// Each thread transposes 4 particles: 8 coalesced b128 loads, 8 b128 stores.
// ---------------------------------------------------------------------------
__global__ __launch_bounds__(256) void pack_kernel(const float* __restrict__ feat,
                                                   float* __restrict__ packed)
{
    const int N4 = NPART / 4;
    const int p4 = blockIdx.x * blockDim.x + threadIdx.x;
    if (p4 >= N4) return;

    const v4f* f = (const v4f*)feat;
    v4f r[BC];
#pragma unroll
    for (int k = 0; k < BC; ++k)
        r[k] = f[(size_t)k * N4 + p4];

    v4f* o = (v4f*)packed;
#pragma unroll
    for (int j = 0; j < 4; ++j) {
        v4f lo = { r[0][j], r[1][j], r[2][j], r[3][j] };
        v4f hi = { r[4][j], r[5][j], r[6][j], r[7][j] };
        const size_t base = ((size_t)p4 * 4 + j) * 2;
        o[base + 0] = lo;
        o[base + 1] = hi;
    }
}

// ---------------------------------------------------------------------------
// Gather body: two voxels (v, v+1), 32B record each from packed table,
// 8 planes x one float2 non-temporal store (output never re-read -> TH=NT,
// keeps the packed table resident in L2).
// ---------------------------------------------------------------------------
__device__ __forceinline__ void gather_store_pair(const float* __restrict__ packed,
                                                  float* __restrict__ out,
                                                  long long i0, long long i1,
                                                  size_t v)
{
    const v4f* p0 = (const v4f*)(packed + ((size_t)i0 << 3));
    const v4f* p1 = (const v4f*)(packed + ((size_t)i1 << 3));
    v4f a0 = p0[0], a1 = p0[1];
    v4f b0 = p1[0], b1 = p1[1];
#pragma unroll
    for (int k = 0; k < 4; ++k) {
        v2f wlo = { a0[k], b0[k] };
        __builtin_nontemporal_store(wlo, (v2f*)(out + (size_t)k * VTOT + v));
        v2f whi = { a1[k], b1[k] };
        __builtin_nontemporal_store(whi, (v2f*)(out + (size_t)(k + 4) * VTOT + v));
    }
}

// ---------------------------------------------------------------------------
// Pass 2: gather. 256 thr/block, 2 voxels/thread, 4 tiles/block (512 vox/tile).
// Index stream staged to LDS with the CDNA5 async engine, double-buffered:
// tile t+1's indices fly in via ASYNCcnt while tile t's scattered gathers run.
// Each thread reads back only the 16B it copied -> no workgroup barrier.
// ---------------------------------------------------------------------------
#define GATHER_BLOCKS      8192
#define TILES_PER_BLOCK    4
#define TILE_VOX           512   // 256 threads * 2 voxels

__global__ __launch_bounds__(256) void gather_packed_kernel(
    const long long* __restrict__ idx,
    const float* __restrict__ packed,
    float* __restrict__ out)
{
    const int tid = threadIdx.x;
#if USE_ASYNC_LDS
    __shared__ long long sidx[2][TILE_VOX];
    const int firstTile = blockIdx.x * TILES_PER_BLOCK;

    {   // prologue: kick tile 0 into buffer 0
        long long* g = const_cast<long long*>(idx) +
                       (size_t)firstTile * TILE_VOX + 2 * tid;
        __builtin_amdgcn_global_load_async_to_lds_b128(
            (as1_v4i*)g, (as3_v4i*)&sidx[0][2 * tid], 0, 0);
    }
#pragma unroll
    for (int it = 0; it < TILES_PER_BLOCK; ++it) {
        const int buf = it & 1;
        if (it + 1 < TILES_PER_BLOCK) {
            long long* g = const_cast<long long*>(idx) +
                           (size_t)(firstTile + it + 1) * TILE_VOX + 2 * tid;
            __builtin_amdgcn_global_load_async_to_lds_b128(
                (as1_v4i*)g, (as3_v4i*)&sidx[buf ^ 1][2 * tid], 0, 0);
            __builtin_amdgcn_s_wait_asynccnt(1);   // current tile landed
        } else {
            __builtin_amdgcn_s_wait_asynccnt(0);
        }
        asm volatile("" ::: "memory");             // keep LDS read after wait

        v2ll ii = *(const v2ll*)&sidx[buf][2 * tid];
        const size_t v = (size_t)(firstTile + it) * TILE_VOX + 2 * tid;
        gather_store_pair(packed, out, ii[0], ii[1], v);
    }
#else
    // Fallback: register-pipelined NT index loads (idx streamed once -> NT).
    const int firstTile = blockIdx.x * TILES_PER_BLOCK;
#pragma unroll
    for (int it = 0; it < TILES_PER_BLOCK; ++it) {
        const size_t v = (size_t)(firstTile + it) * TILE_VOX + 2 * tid;
        v2ll ii = __builtin_nontemporal_load((const v2ll*)(idx + v));
        gather_store_pair(packed, out, ii[0], ii[1], v);
    }
#endif
}

// ---------------------------------------------------------------------------
// Fallback if workspace is too small for the 64 MiB packed table:
// direct strided gather (8 scattered 4B reads / voxel-pair column).
// ---------------------------------------------------------------------------
__global__ __launch_bounds__(256) void gather_direct_kernel(
    const long long* __restrict__ idx,
    const float* __restrict__ feat,
    float* __restrict__ out)
{
    const size_t v = ((size_t)blockIdx.x * blockDim.x + threadIdx.x) * 2;
    if (v >= (size_t)VTOT) return;
    v2ll ii = __builtin_nontemporal_load((const v2ll*)(idx + v));
#pragma unroll
    for (int k = 0; k < BC; ++k) {
        v2f w = { feat[(size_t)k * NPART + ii[0]],
                  feat[(size_t)k * NPART + ii[1]] };
        __builtin_nontemporal_store(w, (v2f*)(out + (size_t)k * VTOT + v));
    }
}

// ---------------------------------------------------------------------------
extern "C" void kernel_launch(void* const* d_in, const int* in_sizes, int n_in,
                              void* d_out, int out_size, void* d_ws, size_t ws_size,
                              hipStream_t stream)
{
    const float*     feat = (const float*)d_in[0];       // (B,C,N) f32
    const long long* idx  = (const long long*)d_in[1];   // (Z,Y,X) i64
    // d_in[2] = level_deltas : unused by the reference output
    float* out = (float*)d_out;                          // (B,C,Z,Y,X) f32

    const size_t packedBytes = (size_t)NPART * BC * sizeof(float); // 64 MiB
    if (ws_size >= packedBytes) {
        float* packed = (float*)d_ws;
        pack_kernel<<<(NPART / 4 + 255) / 256, 256, 0, stream>>>(feat, packed);
        gather_packed_kernel<<<GATHER_BLOCKS, 256, 0, stream>>>(idx, packed, out);
    } else {
        gather_direct_kernel<<<VTOT / 2 / 256, 256, 0, stream>>>(idx, feat, out);
    }
}